// TriggerGenerator_1597727834313
// MI455X (gfx1250) — compile-verified
//
#include <hip/hip_runtime.h>
#include <cstddef>

// ---------------------------------------------------------------------------
// GCN 2-layer forward for MI455X (gfx1250, wave32).
//
//   Per layer: H = X @ W (WMMA f32 16x16x4, exact fp32)
//              out[d] = act( sum_{s in inedges(d)} H[s]*dinv[s]*dinv[d]
//                            + H[d]*dinv[d]^2 + bias )
//
// Aggregation is gather-based over a dst-CSR built once per call:
//   no float atomics, register accumulation, fused self-loop/bias/activation.
// H (51 MB) is L2-resident (192 MB L2) -> random row gathers hit L2.
// ---------------------------------------------------------------------------

typedef __attribute__((ext_vector_type(2))) float v2f;
typedef __attribute__((ext_vector_type(8))) float v8f;

#define FDIM 128
#define XS_PAD 4      // 132-float LDS row stride -> conflict-free column reads
#define SCAN_B 256    // elements per first-level scan block

// ---- degree count / normalization -----------------------------------------
__global__ __launch_bounds__(256) void k_zero_i32(int* __restrict__ p, int n) {
  int i = blockIdx.x * blockDim.x + threadIdx.x;
  if (i < n) p[i] = 0;
}

__global__ __launch_bounds__(256) void k_count(const long long* __restrict__ dst,
                                               int* __restrict__ degi, int e) {
  int i = blockIdx.x * blockDim.x + threadIdx.x;
  if (i < e) atomicAdd(&degi[(int)dst[i]], 1);
}

__global__ __launch_bounds__(256) void k_dinv(const int* __restrict__ degi,
                                              float* __restrict__ dinv, int n) {
  int i = blockIdx.x * blockDim.x + threadIdx.x;
  if (i < n) dinv[i] = rsqrtf(1.0f + (float)degi[i]);  // +1 = self-loop
}

// ---- two-level exclusive scan of degi -> rowptr ----------------------------
__global__ __launch_bounds__(SCAN_B) void k_scan_block(const int* __restrict__ in,
                                                       int* __restrict__ out,
                                                       int* __restrict__ sums, int n) {
  __shared__ int tmp[SCAN_B];
  const int t = threadIdx.x;
  const int i = blockIdx.x * SCAN_B + t;
  const int v = (i < n) ? in[i] : 0;
  tmp[t] = v;
  __syncthreads();
  for (int off = 1; off < SCAN_B; off <<= 1) {
    int add = (t >= off) ? tmp[t - off] : 0;
    __syncthreads();
    tmp[t] += add;
    __syncthreads();
  }
  if (i < n) out[i] = tmp[t] - v;              // exclusive
  if (t == SCAN_B - 1) sums[blockIdx.x] = tmp[t];
}

__global__ __launch_bounds__(1024) void k_scan_top(int* __restrict__ sums, int nb) {
  __shared__ int tmp[1024];
  const int t = threadIdx.x;
  const int v = (t < nb) ? sums[t] : 0;
  tmp[t] = v;
  __syncthreads();
  for (int off = 1; off < 1024; off <<= 1) {
    int add = (t >= off) ? tmp[t - off] : 0;
    __syncthreads();
    tmp[t] += add;
    __syncthreads();
  }
  if (t < nb) sums[t] = tmp[t] - v;            // exclusive
}

__global__ __launch_bounds__(SCAN_B) void k_scan_add(int* __restrict__ rowptr,
                                                     const int* __restrict__ sums,
                                                     int* __restrict__ cursor, int n) {
  const int i = blockIdx.x * SCAN_B + threadIdx.x;
  if (i < n) {
    const int v = rowptr[i] + sums[blockIdx.x];
    rowptr[i] = v;
    cursor[i] = v;
  }
}

// ---- CSR fill: csr_src[slot(dst)] = src ------------------------------------
__global__ __launch_bounds__(256) void k_fill(const long long* __restrict__ src,
                                              const long long* __restrict__ dst,
                                              int* __restrict__ cursor,
                                              int* __restrict__ csr_src, int e) {
  int i = blockIdx.x * blockDim.x + threadIdx.x;
  if (i < e) {
    const int d = (int)dst[i];
    const int pos = atomicAdd(&cursor[d], 1);
    csr_src[pos] = (int)src[i];
  }
}

// ---- dense transform: H[row0..row0+15, :] = X @ W  (W is 128x128) ----------
// 256 threads = 8 waves; wave w computes the 16x16 tile at columns [16w,16w+16).
// K-loop: 32 x V_WMMA_F32_16X16X4_F32.
__global__ __launch_bounds__(256) void k_gemm(const float* __restrict__ X,
                                              const float* __restrict__ W,
                                              float* __restrict__ H, int nrows) {
  __shared__ float Xs[16][FDIM + XS_PAD];

  const int tid  = threadIdx.x;
  const int row0 = blockIdx.x * 16;

  {
    const int base = tid * 8;
    const int r = base >> 7;
    const int c = base & (FDIM - 1);
    float4 u0 = {0.f, 0.f, 0.f, 0.f};
    float4 u1 = {0.f, 0.f, 0.f, 0.f};
    if (row0 + r < nrows) {
      const float* p = X + (size_t)(row0 + r) * FDIM + c;
      u0 = *(const float4*)(p);
      u1 = *(const float4*)(p + 4);
    }
    *(float4*)&Xs[r][c]     = u0;
    *(float4*)&Xs[r][c + 4] = u1;
  }
  __syncthreads();

  const int wave    = tid >> 5;
  const int lane    = tid & 31;
  const int half    = lane >> 4;     // K pair select / M offset
  const int ln      = lane & 15;     // A: row M ; B/D: column N
  const int colBase = wave * 16;

  v8f acc = {};
  #pragma unroll
  for (int k0 = 0; k0 < FDIM; k0 += 4) {
    const int ka = k0 + 2 * half;
    v2f a, b;
    a.x = Xs[ln][ka];
    a.y = Xs[ln][ka + 1];
    b.x = W[(size_t)ka * FDIM + colBase + ln];
    b.y = W[(size_t)(ka + 1) * FDIM + colBase + ln];
    acc = __builtin_amdgcn_wmma_f32_16x16x4_f32(
        false, a, false, b, (short)0, acc, false, false);
  }

  float* dst = H + (size_t)row0 * FDIM + colBase + ln;
  #pragma unroll
  for (int v = 0; v < 8; ++v) {
    const int m = v + 8 * half;
    if (row0 + m < nrows) dst[(size_t)m * FDIM] = acc[v];
  }
}

// ---- fused aggregation + self-loop + bias + activation ---------------------
// One wave per node; lane j owns columns [4j, 4j+4). Edge indices batched:
// each lane loads one (src, dinv[src]) pair, __shfl broadcasts per edge.
// ACT: 0 = ReLU (layer 1), 1 = sigmoid + owner columns (layer 2).
template <int ACT>
__global__ __launch_bounds__(256) void k_agg(const float* __restrict__ H,
                                             const int* __restrict__ rowptr,
                                             const int* __restrict__ cnt,
                                             const int* __restrict__ csr_src,
                                             const float* __restrict__ dinv,
                                             const float* __restrict__ bias,
                                             float* __restrict__ out, int n) {
  const int node = (int)(((long long)blockIdx.x * blockDim.x + threadIdx.x) >> 5);
  const int lane = threadIdx.x & 31;
  if (node >= n) return;                      // wave-uniform

  const int   start = rowptr[node];
  const int   c     = cnt[node];
  const float dd    = dinv[node];
  const int   c4    = lane * 4;

  // self-loop + bias
  const float4 hs = *(const float4*)(H + (size_t)node * FDIM + c4);
  const float4 bb = *(const float4*)(bias + c4);
  const float  sn = dd * dd;
  float4 acc;
  acc.x = hs.x * sn + bb.x;
  acc.y = hs.y * sn + bb.y;
  acc.z = hs.z * sn + bb.z;
  acc.w = hs.w * sn + bb.w;

  for (int kb = 0; kb < c; kb += 32) {
    const int nk = (c - kb < 32) ? (c - kb) : 32;
    int   sidx = 0;
    float nrm  = 0.0f;
    if (lane < nk) {
      sidx = csr_src[start + kb + lane];
      nrm  = dinv[sidx] * dd;
    }
    for (int j = 0; j < nk; ++j) {
      const int   s = __shfl(sidx, j, 32);
      const float w = __shfl(nrm, j, 32);
      const float4 h = *(const float4*)(H + (size_t)s * FDIM + c4);
      acc.x += h.x * w;
      acc.y += h.y * w;
      acc.z += h.z * w;
      acc.w += h.w * w;
    }
  }

  float4 o;
  if (ACT == 0) {
    o.x = fmaxf(acc.x, 0.f); o.y = fmaxf(acc.y, 0.f);
    o.z = fmaxf(acc.z, 0.f); o.w = fmaxf(acc.w, 0.f);
  } else {
    o.x = 1.0f / (1.0f + __expf(-acc.x));
    o.y = 1.0f / (1.0f + __expf(-acc.y));
    o.z = 1.0f / (1.0f + __expf(-acc.z));
    o.w = 1.0f / (1.0f + __expf(-acc.w));
    if (c4 + 0 >= FDIM - 5) o.x = 0.7f;
    if (c4 + 1 >= FDIM - 5) o.y = 0.7f;
    if (c4 + 2 >= FDIM - 5) o.z = 0.7f;
    if (c4 + 3 >= FDIM - 5) o.w = 0.7f;
  }
  *(float4*)(out + (size_t)node * FDIM + c4) = o;
}

// ---------------------------------------------------------------------------
extern "C" void kernel_launch(void* const* d_in, const int* in_sizes, int n_in,
                              void* d_out, int out_size, void* d_ws, size_t ws_size,
                              hipStream_t stream) {
  const float*     x   = (const float*)d_in[0];
  const long long* ei  = (const long long*)d_in[1];   // int64 [2, E]
  const float*     W1  = (const float*)d_in[2];
  const float*     b1  = (const float*)d_in[3];
  const float*     W2  = (const float*)d_in[4];
  const float*     b2  = (const float*)d_in[5];
  float*           out = (float*)d_out;

  const int N = in_sizes[0] / FDIM;
  const int E = in_sizes[1] / 2;
  const long long* src = ei;
  const long long* dst = ei + E;

  // Workspace layout (aligned 256B): dinv[N] f32 | degi[N] | rowptr[N] |
  // cursor[N] | sums[1024] i32 | csr_src[E] i32 | bufH[N*128] | bufA[N*128]
  char* p = (char*)d_ws;
  auto carve = [&](size_t bytes) {
    char* q = p;
    p += (bytes + 255) & ~(size_t)255;
    return q;
  };
  float* dinv    = (float*)carve((size_t)N * 4);
  int*   degi    = (int*)  carve((size_t)N * 4);
  int*   rowptr  = (int*)  carve((size_t)N * 4);
  int*   cursor  = (int*)  carve((size_t)N * 4);
  int*   sums    = (int*)  carve(1024 * 4);
  int*   csr_src = (int*)  carve((size_t)E * 4);
  float* bufH    = (float*)carve((size_t)N * FDIM * 4);
  float* bufA    = (float*)carve((size_t)N * FDIM * 4);

  const int T  = 256;
  const int gN = (N + T - 1) / T;
  const int gE = (E + T - 1) / T;
  const int nb1 = (N + SCAN_B - 1) / SCAN_B;          // <= 1024 for N <= 262144
  const int gGemm = (N + 15) / 16;
  const int gAgg  = (int)(((long long)N * 32 + T - 1) / T);

  // Degree + normalization
  k_zero_i32<<<gN, T, 0, stream>>>(degi, N);
  k_count   <<<gE, T, 0, stream>>>(dst, degi, E);
  k_dinv    <<<gN, T, 0, stream>>>(degi, dinv, N);

  // CSR by destination: rowptr = exclusive_scan(degi); fill csr_src
  k_scan_block<<<nb1, SCAN_B, 0, stream>>>(degi, rowptr, sums, N);
  k_scan_top  <<<1, 1024, 0, stream>>>(sums, nb1);
  k_scan_add  <<<nb1, SCAN_B, 0, stream>>>(rowptr, sums, cursor, N);
  k_fill      <<<gE, T, 0, stream>>>(src, dst, cursor, csr_src, E);

  // Layer 1: H1 = X @ W1 ; A1 = relu(gather-agg + self + b1)
  k_gemm  <<<gGemm, T, 0, stream>>>(x, W1, bufH, N);
  k_agg<0><<<gAgg, T, 0, stream>>>(bufH, rowptr, degi, csr_src, dinv, b1, bufA, N);

  // Layer 2: H2 = A1 @ W2 ; out = sigmoid(gather-agg + self + b2), owner cols
  k_gemm  <<<gGemm, T, 0, stream>>>(bufA, W2, bufH, N);
  k_agg<1><<<gAgg, T, 0, stream>>>(bufH, rowptr, degi, csr_src, dinv, b2, out, N);
}